// DenseRadiusGraph_8933531976109
// MI455X (gfx1250) — compile-verified
//
#include <hip/hip_runtime.h>
#include <hip/hip_bf16.h>
#include <math.h>

// Problem constants (match reference)
#define CUTOFF   10.0f
#define KNBR     32
#define BGR      16
#define NNODE    2048

#define WAVES_PER_BLOCK 2
#define THREADS         (WAVES_PER_BLOCK * 32)
#define ROWS_PER_BLOCK  (WAVES_PER_BLOCK * 32)   // one row per thread

typedef __attribute__((ext_vector_type(2))) float        v2f;
typedef __attribute__((ext_vector_type(8))) float        v8f;
typedef __attribute__((ext_vector_type(4))) unsigned int u32x4;
typedef __attribute__((ext_vector_type(8))) int          i32x8;
typedef __attribute__((ext_vector_type(4))) int          i32x4;

__global__ __launch_bounds__(THREADS)
void radius_graph_wmma(const float* __restrict__ pos,
                       int* __restrict__ erow, int* __restrict__ ecol,
                       float* __restrict__ ew, float* __restrict__ evld)
{
    __shared__ float posL[NNODE * 3];                    // whole graph's positions (24KB)
    __shared__ float sq[NNODE];                          // |p|^2 per node
    __shared__ float dotT[WAVES_PER_BLOCK][32][16];      // per-wave 32x16 dot tile
    __shared__ float listD[ROWS_PER_BLOCK][KNBR + 1];    // per-row sorted dists (+pad)
    __shared__ int   listI[ROWS_PER_BLOCK][KNBR + 1];    // per-row neighbor ids

    const int tid  = threadIdx.x;
    const int lane = tid & 31;
    const int wv   = tid >> 5;
    const int b    = blockIdx.x / (NNODE / ROWS_PER_BLOCK);
    const int rb   = blockIdx.x % (NNODE / ROWS_PER_BLOCK);
    const int gb   = b * NNODE;                       // first global node of graph
    const int row0 = rb * ROWS_PER_BLOCK + wv * 32;   // wave's first row (graph-local)
    const int iLoc = row0 + lane;                     // this lane's row (graph-local)

    // ---- TDM: DMA this graph's 6144 floats of positions into LDS ----
    // D# group0: count=1 | lds_addr | global_addr[56:0] | type=2
    // D# group1: data_size=4B, tensor_dim0 = tile_dim0 = 6144, stride0 = 6144,
    //            tensor_dim1 = tile_dim1 = 1 (single row), no pad/iterate/multicast.
    if (wv == 0) {
        const unsigned long long ga =
            (unsigned long long)(const void*)(pos + 3ull * (unsigned long long)gb);
        const unsigned int ldsOff =
            (unsigned int)(unsigned long long)(const void*)&posL[0];
        u32x4 g0;
        g0[0] = 1u;                                              // count=1, user descriptor
        g0[1] = ldsOff;                                          // lds_addr
        g0[2] = (unsigned int)(ga & 0xFFFFFFFFull);              // global_addr[31:0]
        g0[3] = (unsigned int)((ga >> 32) & 0x01FFFFFFull)       // global_addr[56:32]
              | (2u << 30);                                      // type = 2 ("image")
        i32x8 g1;
        g1[0] = (int)(2u << 16);        // workgroup_mask=0 | data_size=2 (4 bytes)
        g1[1] = (int)(6144u << 16);     // tensor_dim0[15:0]=6144 (bits 63:48)
        g1[2] = (int)(1u << 16);        // tensor_dim0[31:16]=0 | tensor_dim1[15:0]=1
        g1[3] = (int)(6144u << 16);     // tensor_dim1[31:16]=0 | tile_dim0=6144
        g1[4] = 1;                      // tile_dim1=1 | tile_dim2=0
        g1[5] = 6144;                   // tensor_dim0_stride[31:0]
        g1[6] = (int)(6144u << 16);     // stride0[47:32]=0 | tensor_dim1_stride[15:0]
        g1[7] = 0;                      // tensor_dim1_stride[47:16]=0
        i32x4 gz4 = {0, 0, 0, 0};       // groups 2/3 unused (<=2D tensor)
        i32x8 gz8 = {0, 0, 0, 0, 0, 0, 0, 0};
        __builtin_amdgcn_tensor_load_to_lds(g0, g1, gz4, gz4, gz8, 0);
        __builtin_amdgcn_s_wait_tensorcnt(0);
    }
    __syncthreads();                    // posL[] ready for all waves

    // ---- |p|^2 table from LDS positions ----
    for (int n = tid; n < NNODE; n += THREADS) {
        float x = posL[3 * n + 0];
        float y = posL[3 * n + 1];
        float z = posL[3 * n + 2];
        sq[n] = x * x + y * y + z * z;
    }

    // ---- own position (from LDS) ----
    const float px  = posL[3 * iLoc + 0];
    const float py  = posL[3 * iLoc + 1];
    const float pz  = posL[3 * iLoc + 2];
    const float sqi = px * px + py * py + pz * pz;

    // ---- A operands (registers, whole kernel) ----
    // V_WMMA_F32_16X16X4_F32 A layout (16x4, MxK):
    //   lanes 0-15 : {K=0, K=1} of row M=lane
    //   lanes 16-31: {K=2, K=3} of row M=lane-16   (K=3 padded with 0)
    v2f a0, a1;
    {
        const int  m  = lane & 15;
        const int  r0 = row0 + m;          // tile 0 row
        const int  r1 = r0 + 16;           // tile 1 row
        const bool lo = (lane < 16);
        a0[0] = lo ? posL[3 * r0 + 0] : posL[3 * r0 + 2];
        a0[1] = lo ? posL[3 * r0 + 1] : 0.0f;
        a1[0] = lo ? posL[3 * r1 + 0] : posL[3 * r1 + 2];
        a1[1] = lo ? posL[3 * r1 + 1] : 0.0f;
    }

    int cnt = 0;
    const float cut2 = CUTOFF * CUTOFF;

    __syncthreads();                       // sq[] ready

    for (int ct = 0; ct < NNODE / 16; ++ct) {
        const int c0 = ct * 16;

        // B operand (4x16, KxN): B[k][n] = posL[col n][k]  -- all from LDS now
        v2f bv;
        {
            const int  n  = c0 + (lane & 15);
            const bool lo = (lane < 16);
            bv[0] = lo ? posL[3 * n + 0] : posL[3 * n + 2];
            bv[1] = lo ? posL[3 * n + 1] : 0.0f;
        }

        // Two 16x16 gram tiles: D = A x B (f32, exact)
        v8f cz = {};
        v8f d0 = __builtin_amdgcn_wmma_f32_16x16x4_f32(
                     false, a0, false, bv, (short)0, cz, false, false);
        v8f d1 = __builtin_amdgcn_wmma_f32_16x16x4_f32(
                     false, a1, false, bv, (short)0, cz, false, false);

        // Scatter D tiles to LDS.
        // D layout: lane l, VGPR v -> row (v + 8*(l>=16)), col (l & 15)
        {
            const int rofs = (lane < 16) ? 0 : 8;
            const int col  = lane & 15;
            #pragma unroll
            for (int v = 0; v < 8; ++v) {
                dotT[wv][v + rofs][col]      = d0[v];
                dotT[wv][16 + v + rofs][col] = d1[v];
            }
        }
        __syncthreads();   // tile visible

        // Selection: lane owns row `lane` of this wave's 32-row strip
        #pragma unroll 4
        for (int c = 0; c < 16; ++c) {
            const int   j   = c0 + c;
            const float dot = dotT[wv][lane][c];
            float d2 = sqi + sq[j] - 2.0f * dot;
            d2 = fmaxf(d2, 0.0f);
            if ((d2 <= cut2) && (j != iLoc)) {
                const float d = sqrtf(d2);
                if (cnt < KNBR || d < listD[tid][KNBR - 1]) {
                    int p = (cnt < KNBR) ? cnt : (KNBR - 1);
                    while (p > 0 && listD[tid][p - 1] > d) {
                        listD[tid][p] = listD[tid][p - 1];
                        listI[tid][p] = listI[tid][p - 1];
                        --p;
                    }
                    listD[tid][p] = d;
                    listI[tid][p] = j;
                    if (cnt < KNBR) ++cnt;
                }
            }
        }
        __syncthreads();   // done reading tile before next iteration overwrites
    }

    // ---- emit padded fixed-shape edge list for this row ----
    const int  gi    = gb + iLoc;                // global row node index
    const long ebase = (long)gi * KNBR;
    for (int k = 0; k < KNBR; ++k) {
        const bool v    = (k < cnt);
        const int  jloc = v ? listI[tid][k] : 0;
        float wgt = 0.0f;
        if (v) {   // recompute weight exactly from positions, like the reference
            const float dx = px - posL[3 * jloc + 0];
            const float dy = py - posL[3 * jloc + 1];
            const float dz = pz - posL[3 * jloc + 2];
            wgt = sqrtf(dx * dx + dy * dy + dz * dz);
        }
        erow[ebase + k] = v ? gi : 0;
        ecol[ebase + k] = v ? (gb + jloc) : 0;
        ew  [ebase + k] = wgt;
        evld[ebase + k] = v ? 1.0f : 0.0f;
    }
}

extern "C" void kernel_launch(void* const* d_in, const int* in_sizes, int n_in,
                              void* d_out, int out_size, void* d_ws, size_t ws_size,
                              hipStream_t stream)
{
    (void)in_sizes; (void)n_in; (void)out_size; (void)d_ws; (void)ws_size;

    const float* pos = (const float*)d_in[0];
    // d_in[1] = batch (int32); graphs are equal-size and contiguous -> unused.

    const long E = (long)BGR * NNODE * KNBR;
    int*   erow = (int*)d_out;                 // edge_index[0]  (int32 bits)
    int*   ecol = erow + E;                    // edge_index[1]
    float* ew   = (float*)d_out + 2 * E;       // edge weights
    float* evld = (float*)d_out + 3 * E;       // valid mask as 0/1 floats

    const dim3 grid(BGR * NNODE / ROWS_PER_BLOCK);   // 512 blocks
    radius_graph_wmma<<<grid, THREADS, 0, stream>>>(pos, erow, ecol, ew, evld);
}